// SequentialAttentionCountingModel_78597901517451
// MI455X (gfx1250) — compile-verified
//
#include <hip/hip_runtime.h>
#include <hip/hip_bf16.h>
#include <math.h>

// ---------------------------------------------------------------------------
// SequentialAttentionCountingModel for MI455X (gfx1250, wave32).
//
// All large linears are [16,2048] x [N,2048]^T GEMMs -> V_WMMA_F32_16X16X4_F32
// with M=16 == batch.  Memory-bound on fp32 weight streaming (~2.6 GB over the
// 6 steps -> ~115us at 23.3 TB/s), so:
//   * f32 WMMA at native precision (no conversion saves no bytes),
//   * double-buffered GLOBAL_LOAD_ASYNC_TO_LDS_B128 staging (ASYNCcnt) to
//     overlap the HBM weight stream with the WMMA chain,
//   * weights streamed with TH=NT so the L2 keeps `spatial` (134 MB, re-read
//     24x by the fovea einsum) resident instead of being thrashed.
//
// Input flattening assumption (depth-first, dict insertion order):
//  0 features[16,2048] 1 spatial[16,2048,32,32]
//  2..5 qw,kw,vw,ow[2048,2048]  6..9 qb,kb,vb,ob[2048]
//  10+8i : reason block i: w1,b1,g1,bt1,w2,b2,g2,bt2 (i=0..2)
//  34 cw1 35 cb1 36 cw2 37 cb2 38 sw1 39 sb1 40 sw2 41 sb2
//  42..45 l0: wih,whh,bih,bhh   46..49 l1: wih,whh,bih,bhh
//  50 cnt_w1 51 cnt_b1 52 cnt_w2 53 cnt_b2
// ---------------------------------------------------------------------------

typedef float v2f __attribute__((ext_vector_type(2)));
typedef float v8f __attribute__((ext_vector_type(8)));
typedef int   v4i_ __attribute__((__vector_size__(16)));

#define D_     2048
#define B_     16
#define KC     128
#define LDX    (KC + 4)

#define AS1 __attribute__((address_space(1)))
#define AS3 __attribute__((address_space(3)))

#if defined(__gfx1250__) && __has_builtin(__builtin_amdgcn_global_load_async_to_lds_b128)
#define USE_ASYNC_BUILTIN 1
#elif defined(__gfx1250__)
#define USE_ASYNC_ASM 1
#endif

// 16-byte global->LDS copy. RT flavor (activations: reused by many blocks).
__device__ __forceinline__ void cp16_rt(float* lds, const float* g) {
#if defined(USE_ASYNC_BUILTIN)
    __builtin_amdgcn_global_load_async_to_lds_b128(
        (AS1 v4i_*)(unsigned long long)g,
        (AS3 v4i_*)(unsigned)(unsigned long long)lds, 0, 0);
#elif defined(USE_ASYNC_ASM)
    unsigned lo = (unsigned)(unsigned long long)lds;
    asm volatile("global_load_async_to_lds_b128 %0, %1, off"
                 :: "v"(lo), "v"(g) : "memory");
#else
    *(float4*)lds = *(const float4*)g;
#endif
}

// NT flavor (weights: streamed once per step, keep them out of L2 so the
// 134 MB `spatial` tensor stays resident for the fovea einsum).
__device__ __forceinline__ void cp16_nt(float* lds, const float* g) {
#if defined(USE_ASYNC_BUILTIN)
    __builtin_amdgcn_global_load_async_to_lds_b128(
        (AS1 v4i_*)(unsigned long long)g,
        (AS3 v4i_*)(unsigned)(unsigned long long)lds, 0, /*cpol TH=NT*/1);
#elif defined(USE_ASYNC_ASM)
    unsigned lo = (unsigned)(unsigned long long)lds;
    asm volatile("global_load_async_to_lds_b128 %0, %1, off th:TH_LOAD_NT"
                 :: "v"(lo), "v"(g) : "memory");
#else
    *(float4*)lds = *(const float4*)g;
#endif
}

__device__ __forceinline__ void stage_wait() {
#if defined(USE_ASYNC_BUILTIN) || defined(USE_ASYNC_ASM)
#if __has_builtin(__builtin_amdgcn_s_wait_asynccnt)
    __builtin_amdgcn_s_wait_asynccnt(0);
#else
    asm volatile("s_wait_asynccnt 0" ::: "memory");
#endif
#endif
}

// ------------------------- GEMM: out[16,N] = X[16,K] @ W[N,K]^T -------------
// 4 waves / block, each wave owns one 16x16 output tile (64 N-columns/block).
// Double-buffered async LDS staging; WMMA f32 16x16x4 over K chunks of 128.
__global__ __launch_bounds__(128) void gemm16_wmma(
    const float* __restrict__ X, const float* __restrict__ W,
    const float* __restrict__ bias, const float* __restrict__ res,
    float* __restrict__ out, int K, int N, int do_relu)
{
    __shared__ float Xs[2][16 * LDX];
    __shared__ float Ws[2][64 * LDX];

    const int tid  = threadIdx.x;
    const int wave = tid >> 5;
    const int lane = tid & 31;
    const int n0   = blockIdx.x * 64;
    const int kp   = (lane >> 4) << 1;   // 0 or 2
    const int mn   = lane & 15;

    v8f acc = {0.f, 0.f, 0.f, 0.f, 0.f, 0.f, 0.f, 0.f};

    auto stage = [&](int buf, int k0) {
        // X chunk: 16 rows x 128 -> 512 b128 ops, 4 per thread (RT).
        #pragma unroll
        for (int i = 0; i < 4; ++i) {
            int idx = tid + i * 128;
            int r   = idx >> 5;
            int c4  = (idx & 31) << 2;
            cp16_rt(&Xs[buf][r * LDX + c4], X + (size_t)r * K + k0 + c4);
        }
        // W chunk: 64 rows x 128 -> 2048 b128 ops, 16 per thread (NT stream).
        #pragma unroll
        for (int i = 0; i < 16; ++i) {
            int idx = tid + i * 128;
            int r   = idx >> 5;
            int c4  = (idx & 31) << 2;
            cp16_nt(&Ws[buf][r * LDX + c4], W + (size_t)(n0 + r) * K + k0 + c4);
        }
    };

    stage(0, 0);
    int cur = 0;
    for (int k0 = 0; k0 < K; k0 += KC) {
        stage_wait();           // my async copies into buf `cur` have landed
        __syncthreads();        // everyone's copies landed / prev reads done
        if (k0 + KC < K) stage(cur ^ 1, k0 + KC);   // prefetch next chunk

        const float* xrow = &Xs[cur][mn * LDX];
        const float* wrow = &Ws[cur][(wave * 16 + mn) * LDX];
#if __has_builtin(__builtin_amdgcn_wmma_f32_16x16x4_f32)
        #pragma unroll
        for (int kk = 0; kk < KC; kk += 4) {
            v2f a, b;
            a.x = xrow[kk + kp];     a.y = xrow[kk + kp + 1];
            b.x = wrow[kk + kp];     b.y = wrow[kk + kp + 1];
            acc = __builtin_amdgcn_wmma_f32_16x16x4_f32(
                false, a, false, b, (short)0, acc, false, false);
        }
#else
        // Scalar fallback with identical output mapping (host pass only).
        const int mbase_f = (lane >> 4) * 8;
        for (int kk = 0; kk < KC; ++kk) {
            float wv = wrow[kk];
            #pragma unroll
            for (int r = 0; r < 8; ++r)
                acc[r] += Xs[cur][(mbase_f + r) * LDX + kk] * wv;
        }
#endif
        cur ^= 1;
    }

    const int gn    = n0 + wave * 16 + mn;
    const int mbase = (lane >> 4) * 8;      // VGPR r: lanes0-15 M=r, 16-31 M=r+8
    const float bv  = bias ? bias[gn] : 0.f;
    #pragma unroll
    for (int r = 0; r < 8; ++r) {
        int m   = mbase + r;
        float v = acc[r] + bv;
        if (res) v += res[(size_t)m * N + gn];
        if (do_relu) v = fmaxf(v, 0.f);
        out[(size_t)m * N + gn] = v;
    }
}

// ------------------------- LayerNorm over D per row -------------------------
// mode 0: out = relu(LN(in));  mode 1: out += LN(in)
__global__ __launch_bounds__(256) void ln16(
    const float* __restrict__ in, const float* __restrict__ g,
    const float* __restrict__ b, float* __restrict__ out, int mode)
{
    __shared__ float red[256];
    __shared__ float s_m, s_inv;
    const int row = blockIdx.x, tid = threadIdx.x;
    const float* x = in + (size_t)row * D_;

    float s = 0.f;
    for (int i = tid; i < D_; i += 256) s += x[i];
    red[tid] = s; __syncthreads();
    for (int o = 128; o > 0; o >>= 1) { if (tid < o) red[tid] += red[tid + o]; __syncthreads(); }
    if (tid == 0) s_m = red[0] * (1.f / D_);
    __syncthreads();
    const float m = s_m;

    float s2 = 0.f;
    for (int i = tid; i < D_; i += 256) { float d = x[i] - m; s2 += d * d; }
    red[tid] = s2; __syncthreads();
    for (int o = 128; o > 0; o >>= 1) { if (tid < o) red[tid] += red[tid + o]; __syncthreads(); }
    if (tid == 0) s_inv = rsqrtf(red[0] * (1.f / D_) + 1e-5f);
    __syncthreads();
    const float inv = s_inv;

    for (int i = tid; i < D_; i += 256) {
        float v = (x[i] - m) * inv * g[i] + b[i];
        if (mode == 0) out[(size_t)row * D_ + i] = fmaxf(v, 0.f);
        else           out[(size_t)row * D_ + i] += v;
    }
}

// ------------------------- tiny cross-attention core ------------------------
// block = (b,h); 256 threads = HD; T <= 5 tokens.
__global__ __launch_bounds__(256) void attn_core(
    const float* __restrict__ Q, const float* __restrict__ Kc,
    const float* __restrict__ Vc, float* __restrict__ att, int T)
{
    __shared__ float red[256];
    __shared__ float w[8];
    const int b = blockIdx.x >> 3, h = blockIdx.x & 7;
    const int d = threadIdx.x;
    const int base = b * D_ + h * 256;
    const float q = Q[base + d];

    for (int t = 0; t < T; ++t) {
        red[d] = q * Kc[t * (B_ * D_) + base + d];
        __syncthreads();
        for (int o = 128; o > 0; o >>= 1) { if (d < o) red[d] += red[d + o]; __syncthreads(); }
        if (d == 0) w[t] = red[0] * (1.f / 16.f);   // /sqrt(256)
        __syncthreads();
    }
    if (d == 0) {
        float mx = -1e30f;
        for (int t = 0; t < T; ++t) mx = fmaxf(mx, w[t]);
        float s = 0.f;
        for (int t = 0; t < T; ++t) { w[t] = __expf(w[t] - mx); s += w[t]; }
        float inv = 1.f / s;
        for (int t = 0; t < T; ++t) w[t] *= inv;
    }
    __syncthreads();
    float a = 0.f;
    for (int t = 0; t < T; ++t) a += w[t] * Vc[t * (B_ * D_) + base + d];
    att[base + d] = a;
}

// ------------------------- glimpse center/scale + gaussian map --------------
__global__ __launch_bounds__(1024) void glimpse_param(
    const float* __restrict__ u, const float* __restrict__ v,
    const float* __restrict__ cw2, const float* __restrict__ cb2,
    const float* __restrict__ sw2, const float* __restrict__ sb2,
    float* __restrict__ amap)
{
    __shared__ float red[1024];
    __shared__ float cc0, cc1, ss, ssum;
    const int b = blockIdx.x, t = threadIdx.x;
    const float uu = u[b * 1024 + t];

    red[t] = uu * cw2[t]; __syncthreads();
    for (int o = 512; o > 0; o >>= 1) { if (t < o) red[t] += red[t + o]; __syncthreads(); }
    if (t == 0) cc0 = tanhf(red[0] + cb2[0]);
    __syncthreads();

    red[t] = uu * cw2[1024 + t]; __syncthreads();
    for (int o = 512; o > 0; o >>= 1) { if (t < o) red[t] += red[t + o]; __syncthreads(); }
    if (t == 0) cc1 = tanhf(red[0] + cb2[1]);
    __syncthreads();

    red[t] = (t < 512) ? v[b * 512 + t] * sw2[t] : 0.f; __syncthreads();
    for (int o = 512; o > 0; o >>= 1) { if (t < o) red[t] += red[t + o]; __syncthreads(); }
    if (t == 0) {
        float sg = 1.f / (1.f + __expf(-(red[0] + sb2[0])));
        ss = sg * 0.3f + 0.1f;
    }
    __syncthreads();

    const int y = t >> 5, x = t & 31;
    const float fx = -1.f + 2.f * (float)x * (1.f / 31.f);
    const float fy = -1.f + 2.f * (float)y * (1.f / 31.f);
    const float dx = fx - cc0, dy = fy - cc1;
    const float sc = ss + 0.1f;
    const float a  = __expf(-(dx * dx + dy * dy) / (2.f * sc * sc));
    red[t] = a; __syncthreads();
    for (int o = 512; o > 0; o >>= 1) { if (t < o) red[t] += red[t + o]; __syncthreads(); }
    if (t == 0) ssum = red[0];
    __syncthreads();
    amap[b * 1024 + t] = a / (ssum + 1e-8f);
}

// ------------------------- fovea einsum: att[b,d] = <spatial[b,d,:],amap[b]> -
// 8 waves/block, one d per wave; amap staged in LDS; spatial L2-resident.
// float4 loads: 8 VMEM ops per 1024-element row per wave.
__global__ __launch_bounds__(256) void fovea_attend(
    const float* __restrict__ spatial, const float* __restrict__ amap,
    float* __restrict__ sg, float* __restrict__ fov)
{
    __shared__ float am[1024];
    const int bi = blockIdx.x;
    const int b  = bi >> 8;
    const int d0 = (bi & 255) * 8;
    const int tid = threadIdx.x;
    for (int i = tid; i < 256; i += 256)
        *(float4*)&am[i * 4] = *(const float4*)&amap[b * 1024 + i * 4];
    __syncthreads();

    const int wave = tid >> 5, lane = tid & 31;
    const int d = d0 + wave;
    const float4* sp4 = (const float4*)(spatial + ((size_t)b * D_ + d) * 1024);
    const float4* am4 = (const float4*)am;
    float acc = 0.f;
    #pragma unroll
    for (int i = lane; i < 256; i += 32) {
        float4 s4 = sp4[i], a4 = am4[i];
        acc += s4.x * a4.x + s4.y * a4.y + s4.z * a4.z + s4.w * a4.w;
    }
    for (int o = 16; o > 0; o >>= 1) acc += __shfl_down(acc, o, 32);
    if (lane == 0) {
        int idx = b * D_ + d;
        sg[idx]  += acc;
        fov[idx] += acc;
    }
}

// ------------------------- LSTM gate pointwise ------------------------------
__device__ __forceinline__ float sigm_(float x) { return 1.f / (1.f + __expf(-x)); }

__global__ __launch_bounds__(256) void lstm_point(
    const float* __restrict__ G, float* __restrict__ H, float* __restrict__ C,
    float* __restrict__ Hcopy)
{
    const int idx = blockIdx.x * 256 + threadIdx.x;   // 0..32767
    const int b = idx >> 11, d = idx & 2047;
    const float* g = G + (size_t)b * 8192;
    float i  = g[d], f = g[2048 + d], gg = g[4096 + d], o = g[6144 + d];
    float c  = sigm_(f) * C[idx] + sigm_(i) * tanhf(gg);
    float h  = sigm_(o) * tanhf(c);
    C[idx] = c; H[idx] = h;
    if (Hcopy) Hcopy[idx] = h;
}

// ------------------------- elementwise: out = a + b + c*scale (nullable) ----
__global__ void ew_axpy(const float* a, const float* b, const float* c,
                        float scale, float* out, int n)
{
    int i = blockIdx.x * 256 + threadIdx.x;
    if (i >= n) return;
    float v = a ? a[i] : 0.f;
    if (b) v += b[i];
    if (c) v += c[i] * scale;
    out[i] = v;
}

// ------------------------- count head ---------------------------------------
__global__ __launch_bounds__(256) void count_head(
    const float* __restrict__ u, const float* __restrict__ w2,
    const float* __restrict__ b2, float* __restrict__ out)
{
    __shared__ float red[256];
    const int b = blockIdx.x, t = threadIdx.x;
    float s = 0.f;
    for (int i = t; i < 1024; i += 256) s += u[b * 1024 + i] * w2[i];
    red[t] = s; __syncthreads();
    for (int o = 128; o > 0; o >>= 1) { if (t < o) red[t] += red[t + o]; __syncthreads(); }
    if (t == 0) out[b] = fmaxf(red[0] + b2[0], 0.f);
}

// ---------------------------------------------------------------------------
extern "C" void kernel_launch(void* const* d_in, const int* in_sizes, int n_in,
                              void* d_out, int out_size, void* d_ws, size_t ws_size,
                              hipStream_t stream)
{
    (void)in_sizes; (void)n_in; (void)out_size; (void)ws_size;
    const float* features = (const float*)d_in[0];
    const float* spatial  = (const float*)d_in[1];
    const float* qw = (const float*)d_in[2];
    const float* kw = (const float*)d_in[3];
    const float* vw = (const float*)d_in[4];
    const float* ow = (const float*)d_in[5];
    const float* qb = (const float*)d_in[6];
    const float* kb = (const float*)d_in[7];
    const float* vb = (const float*)d_in[8];
    const float* ob = (const float*)d_in[9];
    // reason blocks: base 10 + 8*i
    const float* cw1 = (const float*)d_in[34];
    const float* cb1 = (const float*)d_in[35];
    const float* cw2 = (const float*)d_in[36];
    const float* cb2 = (const float*)d_in[37];
    const float* sw1 = (const float*)d_in[38];
    const float* sb1 = (const float*)d_in[39];
    const float* sw2 = (const float*)d_in[40];
    const float* sb2 = (const float*)d_in[41];
    const float* wih0 = (const float*)d_in[42];
    const float* whh0 = (const float*)d_in[43];
    const float* bih0 = (const float*)d_in[44];
    const float* bhh0 = (const float*)d_in[45];
    const float* wih1 = (const float*)d_in[46];
    const float* whh1 = (const float*)d_in[47];
    const float* bih1 = (const float*)d_in[48];
    const float* bhh1 = (const float*)d_in[49];
    const float* cntw1 = (const float*)d_in[50];
    const float* cntb1 = (const float*)d_in[51];
    const float* cntw2 = (const float*)d_in[52];
    const float* cntb2 = (const float*)d_in[53];

    // workspace layout
    float* p = (float*)d_ws;
    const int BD = B_ * D_;                 // 32768
    float* wsState = p; p += BD;
    float* wsCtx   = p; p += BD;
    float* wsSg    = p; p += BD;
    float* wsFov   = p; p += BD;
    float* wsX     = p; p += BD;
    float* wsHt    = p; p += BD;
    float* wsT     = p; p += BD;
    float* wsA     = p; p += BD;
    float* wsU     = p; p += B_ * 1024;
    float* wsV     = p; p += B_ * 512;
    float* wsAmap  = p; p += B_ * 1024;
    float* wsGate  = p; p += B_ * 8192;
    float* wsH0 = p; p += BD;  float* wsC0 = p; p += BD;
    float* wsH1 = p; p += BD;  float* wsC1 = p; p += BD;
    float* Kc   = p; p += 6 * BD;
    float* Vc   = p; p += 6 * BD;

    auto gemm = [&](const float* X, const float* W, const float* bias,
                    const float* res, float* out, int N, int relu) {
        gemm16_wmma<<<N / 64, 128, 0, stream>>>(X, W, bias, res, out, D_, N, relu);
    };
    auto ew = [&](const float* a, const float* b, const float* c, float s,
                  float* o, int n) {
        ew_axpy<<<(n + 255) / 256, 256, 0, stream>>>(a, b, c, s, o, n);
    };

    // init: state = features; lstm hidden = zeros
    ew(features, nullptr, nullptr, 0.f, wsState, BD);
    ew(nullptr, nullptr, nullptr, 0.f, wsH0, BD);
    ew(nullptr, nullptr, nullptr, 0.f, wsC0, BD);
    ew(nullptr, nullptr, nullptr, 0.f, wsH1, BD);
    ew(nullptr, nullptr, nullptr, 0.f, wsC1, BD);

    for (int t = 0; t < 6; ++t) {
        // ---- cross attention ----
        if (t == 0) {
            ew(nullptr, nullptr, nullptr, 0.f, wsCtx, BD);
        } else {
            gemm(wsState, qw, qb, nullptr, wsT, D_, 0);                 // Q
            attn_core<<<B_ * 8, 256, 0, stream>>>(wsT, Kc, Vc, wsA, t); // softmax*V
            gemm(wsA, ow, ob, nullptr, wsCtx, D_, 0);                   // out proj
        }
        // ---- foveation over state+ctx ----
        ew(wsState, wsCtx, nullptr, 0.f, wsSg, BD);
        ew(nullptr, nullptr, nullptr, 0.f, wsFov, BD);
        for (int g = 0; g < 4; ++g) {
            gemm(wsSg, cw1, cb1, nullptr, wsU, 1024, 1);
            gemm(wsSg, sw1, sb1, nullptr, wsV, 512, 1);
            glimpse_param<<<B_, 1024, 0, stream>>>(wsU, wsV, cw2, cb2, sw2, sb2, wsAmap);
            fovea_attend<<<B_ * 256, 256, 0, stream>>>(spatial, wsAmap, wsSg, wsFov);
        }
        // combined = state + ctx + mean(feats)
        ew(wsState, wsCtx, wsFov, 0.25f, wsX, BD);
        // ---- reasoning: 3 residual MLP blocks ----
        for (int r = 0; r < 3; ++r) {
            const int base = 10 + 8 * r;
            const float* w1  = (const float*)d_in[base + 0];
            const float* b1  = (const float*)d_in[base + 1];
            const float* g1  = (const float*)d_in[base + 2];
            const float* bt1 = (const float*)d_in[base + 3];
            const float* w2  = (const float*)d_in[base + 4];
            const float* b2  = (const float*)d_in[base + 5];
            const float* g2  = (const float*)d_in[base + 6];
            const float* bt2 = (const float*)d_in[base + 7];
            gemm(wsX, w1, b1, nullptr, wsT, D_, 0);
            ln16<<<B_, 256, 0, stream>>>(wsT, g1, bt1, wsHt, 0);   // relu(LN)
            gemm(wsHt, w2, b2, nullptr, wsT, D_, 0);
            ln16<<<B_, 256, 0, stream>>>(wsT, g2, bt2, wsX, 1);    // x += LN
        }
        // ---- LSTM layer 0 ----
        gemm(wsX,  wih0, bih0, nullptr, wsGate, 8192, 0);
        gemm(wsH0, whh0, bhh0, wsGate,  wsGate, 8192, 0);
        lstm_point<<<BD / 256, 256, 0, stream>>>(wsGate, wsH0, wsC0, nullptr);
        // ---- LSTM layer 1 ----
        gemm(wsH0, wih1, bih1, nullptr, wsGate, 8192, 0);
        gemm(wsH1, whh1, bhh1, wsGate,  wsGate, 8192, 0);
        lstm_point<<<BD / 256, 256, 0, stream>>>(wsGate, wsH1, wsC1, wsState);
        // ---- append K,V for the new memory token ----
        gemm(wsState, kw, kb, nullptr, Kc + t * BD, D_, 0);
        gemm(wsState, vw, vb, nullptr, Vc + t * BD, D_, 0);
    }

    // ---- count head: relu(W2 @ relu(W1 @ state + b1) + b2) ----
    gemm(wsState, cntw1, cntb1, nullptr, wsU, 1024, 1);
    count_head<<<B_, 256, 0, stream>>>(wsU, cntw2, cntb2, (float*)d_out);
}